// HiroLinear_36137854828803
// MI455X (gfx1250) — compile-verified
//
#include <hip/hip_runtime.h>

// ---------------------------------------------------------------------------
// HiroLinear on MI455X (gfx1250, wave32, WMMA).
//
//  z0   = leaky(leaky(x0 We1^T + be1) We2^T + be2)            (t=0 only!)
//  Bu_t = u_t B_W^T + B_b                                     (big GEMM)
//  h_t  = h_{t-1} A^T + (A_b + Bu_t), h_{-1}=z0               (chunked scan)
//  y    = (inv_leaky(inv_leaky(z1) Wd1^T + bd1)) Wd2^T + bd2  (two GEMMs)
//
// All GEMMs use V_WMMA_F32_16X16X4_F32 with the weight matrix resident in
// LDS (256KB of the 320KB WGP LDS). Scan parallelized as a 3-phase affine
// scan: local scans (64 WGs) -> carry chain with P=(A^T)^32 (1 WG, 64 steps)
// -> seeded re-scan (64 WGs).
// ---------------------------------------------------------------------------

#define S_DIM   256
#define A_LEN   128
#define IN_DIM  512
#define B_SZ    32
#define T_LEN   2048
#define LCH     32      // chunk length
#define NCH     64      // number of chunks  (LCH*NCH == T_LEN)

typedef float v2f __attribute__((ext_vector_type(2)));
typedef float v8f __attribute__((ext_vector_type(8)));

// ---- wave-level 16x16 f32 WMMA tile: C += Xs[mB:mB+16, 0:K] * Ws[0:K, nB:nB+16]
// Xs: LDS activations [32][xst] row-major (xst even -> 8B aligned v2f loads)
// Ws: LDS weights-as-B [K][wst] row-major (B[k][n] = W[n][k] pre-transposed)
__device__ __forceinline__ v8f wmma_k(const float* Xs, int xst,
                                      const float* Ws, int wst,
                                      int mB, int nB, int K, v8f c) {
  const int lane = threadIdx.x & 31;
  const int g = lane >> 4, ln = lane & 15;
  const float* xrow = Xs + (size_t)(mB + ln) * xst + 2 * g; // A: M=ln, K-pair by half
  const float* wcol = Ws + nB + ln;                         // B: N=ln
  for (int k = 0; k < K; k += 4) {
    v2f a;
    a.x = xrow[k];
    a.y = xrow[k + 1];
    v2f b;
    b.x = wcol[(size_t)(k + 2 * g) * wst];
    b.y = wcol[(size_t)(k + 2 * g + 1) * wst];
    c = __builtin_amdgcn_wmma_f32_16x16x4_f32(false, a, false, b, (short)0, c,
                                              false, false);
  }
  return c;
}

// C/D layout: VGPR j -> row (mB + j + 8*(lane>>4)), col (nB + lane&15)
__device__ __forceinline__ void store_tile(float* dst, size_t rstride,
                                           int mB, int nB, const v8f& c) {
  const int lane = threadIdx.x & 31;
  const int g = lane >> 4, ln = lane & 15;
  for (int j = 0; j < 8; ++j)
    dst[(size_t)(mB + j + 8 * g) * rstride + nB + ln] = c[j];
}

__device__ __forceinline__ float leakyf(float v)     { return v < 0.f ? 0.01f * v : v; }
__device__ __forceinline__ float inv_leakyf(float v) { return v < 0.f ? v / 0.01f : v; }

// ---------------------------------------------------------------------------
// 256x256 transpose: dst[k][n] = src[n][k]  (one-time, tiny)
__global__ void k_transpose256(const float* __restrict__ src, float* __restrict__ dst) {
  const int k = blockIdx.x, n = threadIdx.x;
  dst[(size_t)k * 256 + n] = src[(size_t)n * 256 + k];
}

// ---------------------------------------------------------------------------
// Encoder, t=0 slice only: z0 = leaky(leaky(x0 We1^T+be1) We2^T+be2). 1 WG.
__global__ __launch_bounds__(256, 1) void
k_encoder(const float* __restrict__ pin,
          const float* __restrict__ We1, const float* __restrict__ be1,
          const float* __restrict__ We2, const float* __restrict__ be2,
          float* __restrict__ z0) {
  __shared__ float xs[32][257];
  const int tid = threadIdx.x;
  for (int i = 0; i < 32; ++i) {
    int idx = tid + i * 256;
    int b = idx >> 8, d = idx & 255;
    xs[b][d] = pin[(size_t)b * T_LEN * IN_DIM + d]; // t = 0
  }
  __syncthreads();
  const int e = tid;
  float acc[32];
  // layer 1
  for (int b = 0; b < 32; ++b) acc[b] = be1[e];
  for (int d = 0; d < 256; ++d) {
    float w = We1[(size_t)e * 256 + d];
    for (int b = 0; b < 32; ++b) acc[b] += xs[b][d] * w;
  }
  for (int b = 0; b < 32; ++b) acc[b] = leakyf(acc[b]);
  __syncthreads();
  for (int b = 0; b < 32; ++b) xs[b][e] = acc[b];
  __syncthreads();
  // layer 2
  for (int b = 0; b < 32; ++b) acc[b] = be2[e];
  for (int d = 0; d < 256; ++d) {
    float w = We2[(size_t)e * 256 + d];
    for (int b = 0; b < 32; ++b) acc[b] += xs[b][d] * w;
  }
  for (int b = 0; b < 32; ++b)
    z0[(size_t)b * 256 + e] = leakyf(acc[b]);
}

// ---------------------------------------------------------------------------
// Bu[t][b][:] = u[b][t] B_W^T + B_b, stored (T,B,S). 256 WGs x 8 timesteps.
__global__ __launch_bounds__(256, 1) void
k_bu(const float* __restrict__ pin, const float* __restrict__ BW,
     const float* __restrict__ Bb, float* __restrict__ z1) {
  __shared__ float Ws[128 * 256];   // B[k][n] = B_W[n][k]
  __shared__ float Us[32 * 130];
  __shared__ float Bias[256];
  const int tid = threadIdx.x;
  Bias[tid] = Bb[tid];
  for (int i = 0; i < 128; ++i) {
    int idx = tid + i * 256;        // 256*128 elements
    int n = idx >> 7, k = idx & 127;
    Ws[(size_t)k * 256 + n] = BW[idx];
  }
  const int wave = tid >> 5;
  for (int tt = 0; tt < 8; ++tt) {
    const int t = blockIdx.x * 8 + tt;
    __syncthreads();
    for (int i = 0; i < 16; ++i) {
      int idx = tid + i * 256;
      int b = idx >> 7, k = idx & 127;
      Us[b * 130 + k] = pin[((size_t)b * T_LEN + t) * IN_DIM + S_DIM + k];
    }
    if (tt < 7) { // exercise global_prefetch for next timestep's u-panel
      int b = tid >> 7, k = tid & 127;
      __builtin_prefetch(&pin[((size_t)b * T_LEN + t + 1) * IN_DIM + S_DIM + k], 0, 1);
    }
    __syncthreads();
    for (int i = 0; i < 4; ++i) {
      const int tile = wave * 4 + i;
      const int mB = (tile & 1) * 16, nB = (tile >> 1) * 16;
      const int ln = threadIdx.x & 15;
      float bv = Bias[nB + ln];
      v8f c;
      for (int j = 0; j < 8; ++j) c[j] = bv;
      c = wmma_k(Us, 130, Ws, 256, mB, nB, 128, c);
      store_tile(z1 + (size_t)t * 32 * 256, 256, mB, nB, c);
    }
  }
}

// ---------------------------------------------------------------------------
// dst = src * src (256x256x256) -- used to build P = (A^T)^32. 1 WG.
__global__ __launch_bounds__(256, 1) void
k_sq(const float* __restrict__ src, float* __restrict__ dst) {
  __shared__ float Ps[256 * 256];
  const int tid = threadIdx.x;
  for (int i = 0; i < 256; ++i) Ps[tid + i * 256] = src[tid + i * 256];
  __syncthreads();
  const int wave = tid >> 5;
  for (int i = 0; i < 32; ++i) {
    const int tile = wave * 32 + i;
    const int mB = (tile >> 4) * 16, nB = (tile & 15) * 16;
    v8f c = {0.f, 0.f, 0.f, 0.f, 0.f, 0.f, 0.f, 0.f};
    c = wmma_k(Ps, 256, Ps, 256, mB, nB, 256, c);  // A and B both row-major P
    store_tile(dst, 256, mB, nB, c);
  }
}

// ---------------------------------------------------------------------------
// Phase 1: per-chunk local scan from zero carry; F[chunk] = final state.
__global__ __launch_bounds__(256, 1) void
k_scan_local(const float* __restrict__ AT, const float* __restrict__ Ab,
             const float* __restrict__ Bu, float* __restrict__ F) {
  __shared__ float Bs[256 * 256];   // A^T as B operand: Bs[k][n] = A_W[n][k]
  __shared__ float Hs[32 * 258];
  __shared__ float AbS[256];
  const int tid = threadIdx.x;
  AbS[tid] = Ab[tid];
  for (int i = 0; i < 256; ++i) Bs[tid + i * 256] = AT[tid + i * 256];
  for (int i = 0; i < 32; ++i) {
    int idx = tid + i * 256;
    Hs[(idx >> 8) * 258 + (idx & 255)] = 0.f;
  }
  __syncthreads();
  const int wave = tid >> 5, lane = tid & 31, g = lane >> 4, ln = lane & 15;
  const int chunk = blockIdx.x;
  for (int s = 0; s < LCH; ++s) {
    const int t = chunk * LCH + s;
    const float* but = Bu + (size_t)t * 32 * 256;
    v8f acc[4];
    for (int i = 0; i < 4; ++i) {
      const int tile = wave * 4 + i;
      const int mB = (tile & 1) * 16, nB = (tile >> 1) * 16;
      const float ab = AbS[nB + ln];
      for (int j = 0; j < 8; ++j)
        acc[i][j] = but[(size_t)(mB + j + 8 * g) * 256 + nB + ln] + ab;
      acc[i] = wmma_k(Hs, 258, Bs, 256, mB, nB, 256, acc[i]);
    }
    __syncthreads(); // all waves done reading Hs
    for (int i = 0; i < 4; ++i) {
      const int tile = wave * 4 + i;
      const int mB = (tile & 1) * 16, nB = (tile >> 1) * 16;
      for (int j = 0; j < 8; ++j)
        Hs[(mB + j + 8 * g) * 258 + nB + ln] = acc[i][j];
    }
    __syncthreads();
  }
  for (int i = 0; i < 32; ++i) {
    int idx = tid + i * 256;
    F[(size_t)chunk * 8192 + idx] = Hs[(idx >> 8) * 258 + (idx & 255)];
  }
}

// ---------------------------------------------------------------------------
// Phase 2: sequential carry chain. H[c] = carry into chunk c.
//   H[0] = z0;  H[c+1] = H[c] * P + F[c],  P = (A^T)^LCH.   1 WG, NCH steps.
__global__ __launch_bounds__(256, 1) void
k_chain(const float* __restrict__ z0, const float* __restrict__ F,
        const float* __restrict__ P, float* __restrict__ H) {
  __shared__ float Ps[256 * 256];
  __shared__ float Hs[32 * 258];
  const int tid = threadIdx.x;
  for (int i = 0; i < 256; ++i) Ps[tid + i * 256] = P[tid + i * 256];
  for (int i = 0; i < 32; ++i) {
    int idx = tid + i * 256;
    Hs[(idx >> 8) * 258 + (idx & 255)] = z0[idx];
  }
  __syncthreads();
  const int wave = tid >> 5, lane = tid & 31, g = lane >> 4, ln = lane & 15;
  for (int c = 0; c < NCH; ++c) {
    for (int i = 0; i < 32; ++i) { // store H_c
      int idx = tid + i * 256;
      H[(size_t)c * 8192 + idx] = Hs[(idx >> 8) * 258 + (idx & 255)];
    }
    v8f acc[4];
    for (int i = 0; i < 4; ++i) {
      const int tile = wave * 4 + i;
      const int mB = (tile & 1) * 16, nB = (tile >> 1) * 16;
      for (int j = 0; j < 8; ++j)
        acc[i][j] = F[(size_t)c * 8192 + (size_t)(mB + j + 8 * g) * 256 + nB + ln];
      acc[i] = wmma_k(Hs, 258, Ps, 256, mB, nB, 256, acc[i]);
    }
    __syncthreads();
    for (int i = 0; i < 4; ++i) {
      const int tile = wave * 4 + i;
      const int mB = (tile & 1) * 16, nB = (tile >> 1) * 16;
      for (int j = 0; j < 8; ++j)
        Hs[(mB + j + 8 * g) * 258 + nB + ln] = acc[i][j];
    }
    __syncthreads();
  }
}

// ---------------------------------------------------------------------------
// Phase 3: re-run chunk seeded with H[chunk]; overwrite Bu with z1 in place.
__global__ __launch_bounds__(256, 1) void
k_scan_fix(const float* __restrict__ AT, const float* __restrict__ Ab,
           const float* __restrict__ H, float* __restrict__ z1) {
  __shared__ float Bs[256 * 256];
  __shared__ float Hs[32 * 258];
  __shared__ float AbS[256];
  const int tid = threadIdx.x;
  AbS[tid] = Ab[tid];
  for (int i = 0; i < 256; ++i) Bs[tid + i * 256] = AT[tid + i * 256];
  const int chunk = blockIdx.x;
  for (int i = 0; i < 32; ++i) {
    int idx = tid + i * 256;
    Hs[(idx >> 8) * 258 + (idx & 255)] = H[(size_t)chunk * 8192 + idx];
  }
  __syncthreads();
  const int wave = tid >> 5, lane = tid & 31, g = lane >> 4, ln = lane & 15;
  for (int s = 0; s < LCH; ++s) {
    const int t = chunk * LCH + s;
    float* zt = z1 + (size_t)t * 32 * 256;
    v8f acc[4];
    for (int i = 0; i < 4; ++i) {
      const int tile = wave * 4 + i;
      const int mB = (tile & 1) * 16, nB = (tile >> 1) * 16;
      const float ab = AbS[nB + ln];
      for (int j = 0; j < 8; ++j)   // consume Bu_t (each element read once,
        acc[i][j] = zt[(size_t)(mB + j + 8 * g) * 256 + nB + ln] + ab; // by its owner)
      acc[i] = wmma_k(Hs, 258, Bs, 256, mB, nB, 256, acc[i]);
    }
    __syncthreads();
    for (int i = 0; i < 4; ++i) {
      const int tile = wave * 4 + i;
      const int mB = (tile & 1) * 16, nB = (tile >> 1) * 16;
      for (int j = 0; j < 8; ++j) {
        const float v = acc[i][j];
        Hs[(mB + j + 8 * g) * 258 + nB + ln] = v;
        zt[(size_t)(mB + j + 8 * g) * 256 + nB + ln] = v;  // z1_t over Bu_t
      }
    }
    __syncthreads();
  }
}

// ---------------------------------------------------------------------------
// Decoder layer: Y = inv_leaky(X) * W^T + b.
// permuteOut=0: Y in (T,B,S) layout (may alias X: rows staged in LDS first).
// permuteOut=1: Y = d_out in (B,T,S) layout.
__global__ __launch_bounds__(256, 1) void
k_dec(const float* __restrict__ X, const float* __restrict__ WT,
      const float* __restrict__ bias, float* __restrict__ Y, int permuteOut) {
  __shared__ float Ws[256 * 256];
  __shared__ float Xs[32 * 258];
  __shared__ float Bias[256];
  const int tid = threadIdx.x;
  Bias[tid] = bias[tid];
  for (int i = 0; i < 256; ++i) Ws[tid + i * 256] = WT[tid + i * 256];
  const int wave = tid >> 5, lane = tid & 31, g = lane >> 4, ln = lane & 15;
  for (int tt = 0; tt < 8; ++tt) {
    const int t = blockIdx.x * 8 + tt;
    __syncthreads(); // Ws ready (tt=0) / previous Xs fully consumed
    for (int i = 0; i < 32; ++i) {
      int idx = tid + i * 256;
      float v = X[(size_t)t * 8192 + idx];
      Xs[(idx >> 8) * 258 + (idx & 255)] = inv_leakyf(v);
    }
    __syncthreads();
    for (int i = 0; i < 4; ++i) {
      const int tile = wave * 4 + i;
      const int mB = (tile & 1) * 16, nB = (tile >> 1) * 16;
      const float bv = Bias[nB + ln];
      v8f c;
      for (int j = 0; j < 8; ++j) c[j] = bv;
      c = wmma_k(Xs, 258, Ws, 256, mB, nB, 256, c);
      if (permuteOut) {
        for (int j = 0; j < 8; ++j) {
          const int b = mB + j + 8 * g;
          Y[((size_t)b * T_LEN + t) * 256 + nB + ln] = c[j];
        }
      } else {
        for (int j = 0; j < 8; ++j) {
          const int m = mB + j + 8 * g;
          Y[((size_t)t * 32 + m) * 256 + nB + ln] = c[j];
        }
      }
    }
  }
}

// ---------------------------------------------------------------------------
extern "C" void kernel_launch(void* const* d_in, const int* in_sizes, int n_in,
                              void* d_out, int out_size, void* d_ws, size_t ws_size,
                              hipStream_t stream) {
  const float* pin = (const float*)d_in[0];
  const float* We1 = (const float*)d_in[1];
  const float* be1 = (const float*)d_in[2];
  const float* We2 = (const float*)d_in[3];
  const float* be2 = (const float*)d_in[4];
  const float* A_W = (const float*)d_in[5];
  const float* A_b = (const float*)d_in[6];
  const float* B_W = (const float*)d_in[7];
  const float* B_b = (const float*)d_in[8];
  const float* Wd1 = (const float*)d_in[9];
  const float* bd1 = (const float*)d_in[10];
  const float* Wd2 = (const float*)d_in[11];
  const float* bd2 = (const float*)d_in[12];
  float* out = (float*)d_out;

  float* ws = (float*)d_ws;
  size_t o = 0;
  float* z1  = ws + o; o += (size_t)T_LEN * 32 * 256;  // Bu -> z1 -> y1 (in place)
  float* z0  = ws + o; o += 32 * 256;
  float* F   = ws + o; o += (size_t)NCH * 32 * 256;
  float* H   = ws + o; o += (size_t)NCH * 32 * 256;
  float* AT  = ws + o; o += 256 * 256;
  float* W1T = ws + o; o += 256 * 256;
  float* W2T = ws + o; o += 256 * 256;
  float* Pa  = ws + o; o += 256 * 256;
  float* Pb  = ws + o; o += 256 * 256;

  // Pre-transpose weight matrices once so every GEMM's B operand is row-major.
  k_transpose256<<<256, 256, 0, stream>>>(A_W, AT);
  k_transpose256<<<256, 256, 0, stream>>>(Wd1, W1T);
  k_transpose256<<<256, 256, 0, stream>>>(Wd2, W2T);

  k_encoder<<<1, 256, 0, stream>>>(pin, We1, be1, We2, be2, z0);
  k_bu<<<T_LEN / 8, 256, 0, stream>>>(pin, B_W, B_b, z1);

  // P = (A^T)^32 by repeated squaring
  k_sq<<<1, 256, 0, stream>>>(AT, Pa);  // ^2
  k_sq<<<1, 256, 0, stream>>>(Pa, Pb);  // ^4
  k_sq<<<1, 256, 0, stream>>>(Pb, Pa);  // ^8
  k_sq<<<1, 256, 0, stream>>>(Pa, Pb);  // ^16
  k_sq<<<1, 256, 0, stream>>>(Pb, Pa);  // ^32

  k_scan_local<<<NCH, 256, 0, stream>>>(AT, A_b, z1, F);
  k_chain<<<1, 256, 0, stream>>>(z0, F, Pa, H);
  k_scan_fix<<<NCH, 256, 0, stream>>>(AT, A_b, H, z1);

  k_dec<<<T_LEN / 8, 256, 0, stream>>>(z1, W1T, bd1, z1, 0);   // y1, in place
  k_dec<<<T_LEN / 8, 256, 0, stream>>>(z1, W2T, bd2, out, 1);  // y -> (B,T,S)
}